// Encoder_33775622815757
// MI455X (gfx1250) — compile-verified
//
#include <hip/hip_runtime.h>

// GraphConv x2 + linear head for MI455X (gfx1250, wave32).
//
// Roofline: edge scatter/gather dominates (~2.4 GB -> ~100us @ 23.3 TB/s);
// node features (25.6 MB) are L2-resident (192 MB L2), so fp32 atomic RMW
// stays on-chip. Dense 64x64 GEMMs use exact-fp32 V_WMMA_F32_16X16X4_F32.

typedef float v2f __attribute__((ext_vector_type(2)));
typedef float v8f __attribute__((ext_vector_type(8)));

#define NN 100000
#define NE 1600000
#define D  64

// ---------------------------------------------------------------- zero agg
__global__ __launch_bounds__(256)
void zero4_kernel(float4* __restrict__ p, int n4) {
    int i = blockIdx.x * blockDim.x + threadIdx.x;
    if (i < n4) p[i] = make_float4(0.f, 0.f, 0.f, 0.f);
}

// --------------------------------------------------- edge gather + scatter
// 16 threads per edge; each thread moves a float4 feature chunk.
// 16 consecutive lanes read one contiguous 256B source row (coalesced),
// then issue 4 no-return fp32 global atomic adds into the destination row.
__global__ __launch_bounds__(256)
void scatter_add_kernel(const float* __restrict__ X,
                        const int*   __restrict__ src,
                        const int*   __restrict__ dst,
                        float*       __restrict__ AGG) {
    long long t = (long long)blockIdx.x * blockDim.x + threadIdx.x;
    int e = (int)(t >> 4);
    if (e >= NE) return;
    int f = ((int)t & 15) * 4;
    int s = src[e];
    int d = dst[e];
    const float4 v = *(const float4*)(X + (long long)s * D + f);
    float* o = AGG + (long long)d * D + f;
    __hip_atomic_fetch_add(o + 0, v.x, __ATOMIC_RELAXED, __HIP_MEMORY_SCOPE_AGENT);
    __hip_atomic_fetch_add(o + 1, v.y, __ATOMIC_RELAXED, __HIP_MEMORY_SCOPE_AGENT);
    __hip_atomic_fetch_add(o + 2, v.z, __ATOMIC_RELAXED, __HIP_MEMORY_SCOPE_AGENT);
    __hip_atomic_fetch_add(o + 3, v.w, __ATOMIC_RELAXED, __HIP_MEMORY_SCOPE_AGENT);
}

// ------------------------------------------------------- fused conv (WMMA)
// OUT[n][o] = relu( sum_k AGG[n][k]*Wrel[o][k] + X[n][k]*Wroot[o][k] + b[o] )
// One wave (32 lanes) per 16-node tile; 4 N-tiles of 16 outputs each.
//
// V_WMMA_F32_16X16X4_F32 fragment layouts (ISA 7.12.2):
//   A (16x4):  lane L holds row M=L%16; vgpr r holds K = r + 2*(L/16)
//              -> float2 load at K-base kb + 2*(L/16)
//   B (4x16):  lane L holds col N=L%16; vgpr r holds K = r + 2*(L/16)
//              -> B[k][o] = Wrel[o][k], float2 load at Wrel + o*64 + kbase
//   C/D (16x16): vgpr r, lane L -> (M = r + 8*(L/16), N = L%16)
__global__ __launch_bounds__(32)
void conv_wmma_kernel(const float* __restrict__ X,
                      const float* __restrict__ AGG,
                      const float* __restrict__ Wrel,
                      const float* __restrict__ Wroot,
                      const float* __restrict__ bias,
                      float*       __restrict__ OUT,
                      int do_relu) {
    const int lane = threadIdx.x;      // 0..31, EXEC all-ones (no divergence)
    const int hi   = lane >> 4;        // half-wave select
    const int nl   = lane & 15;
    const int row0 = blockIdx.x * 16;  // 100000 / 16 = 6250 blocks exactly

    // Bias depends only on N -> replicate across all 8 C VGPRs.
    v8f acc[4];
#pragma unroll
    for (int nt = 0; nt < 4; ++nt) {
        float b = bias[nt * 16 + nl];
        acc[nt] = (v8f){b, b, b, b, b, b, b, b};
    }

    const float* arow = AGG + (long long)(row0 + nl) * D;
    const float* xrow = X   + (long long)(row0 + nl) * D;

#pragma unroll 4
    for (int kb = 0; kb < D; kb += 4) {
        const int ka = kb + 2 * hi;
        v2f a = *(const v2f*)(arow + ka);   // AGG fragment (A)
        v2f x = *(const v2f*)(xrow + ka);   // root-input fragment (A)
#pragma unroll
        for (int nt = 0; nt < 4; ++nt) {
            const int o = nt * 16 + nl;
            v2f wr = *(const v2f*)(Wrel  + (long long)o * D + ka);  // B frag
            v2f wo = *(const v2f*)(Wroot + (long long)o * D + ka);  // B frag
            acc[nt] = __builtin_amdgcn_wmma_f32_16x16x4_f32(
                false, a, false, wr, (short)0, acc[nt], false, false);
            acc[nt] = __builtin_amdgcn_wmma_f32_16x16x4_f32(
                false, x, false, wo, (short)0, acc[nt], false, false);
        }
    }

#pragma unroll
    for (int nt = 0; nt < 4; ++nt) {
#pragma unroll
        for (int r = 0; r < 8; ++r) {
            float v = acc[nt][r];
            if (do_relu) v = fmaxf(v, 0.f);
            const int row = row0 + r + 8 * hi;
            OUT[(long long)row * D + nt * 16 + nl] = v;
        }
    }
}

// ------------------------------------------------------------ linear head
// out[n][0..2] = h[n] . W_lin[o] + b_lin[o]  (64 -> 3, trivial VALU work)
__global__ __launch_bounds__(256)
void head_kernel(const float* __restrict__ H,
                 const float* __restrict__ W,
                 const float* __restrict__ b,
                 float*       __restrict__ OUT) {
    int n = blockIdx.x * blockDim.x + threadIdx.x;
    if (n >= NN) return;
    const float4* hr = (const float4*)(H + (long long)n * D);
    float a0 = b[0], a1 = b[1], a2 = b[2];
#pragma unroll
    for (int k4 = 0; k4 < D / 4; ++k4) {
        float4 h  = hr[k4];
        float4 w0 = *(const float4*)(W + 0 * D + k4 * 4);
        float4 w1 = *(const float4*)(W + 1 * D + k4 * 4);
        float4 w2 = *(const float4*)(W + 2 * D + k4 * 4);
        a0 += h.x * w0.x + h.y * w0.y + h.z * w0.z + h.w * w0.w;
        a1 += h.x * w1.x + h.y * w1.y + h.z * w1.z + h.w * w1.w;
        a2 += h.x * w2.x + h.y * w2.y + h.z * w2.z + h.w * w2.w;
    }
    OUT[(long long)n * 3 + 0] = a0;
    OUT[(long long)n * 3 + 1] = a1;
    OUT[(long long)n * 3 + 2] = a2;
}

extern "C" void kernel_launch(void* const* d_in, const int* in_sizes, int n_in,
                              void* d_out, int out_size, void* d_ws, size_t ws_size,
                              hipStream_t stream) {
    const float* x       = (const float*)d_in[0];
    const int*   eidx    = (const int*)  d_in[1];   // [2, NE] row-major int32
    const float* W1_rel  = (const float*)d_in[2];
    const float* b1_rel  = (const float*)d_in[3];
    const float* W1_root = (const float*)d_in[4];
    const float* W2_rel  = (const float*)d_in[5];
    const float* b2_rel  = (const float*)d_in[6];
    const float* W2_root = (const float*)d_in[7];
    const float* W_lin   = (const float*)d_in[8];
    const float* b_lin   = (const float*)d_in[9];
    float* out = (float*)d_out;

    const int* src = eidx;
    const int* dst = eidx + NE;

    // Workspace: agg [NN*D] + h [NN*D] = 51.2 MB
    float* agg = (float*)d_ws;
    float* h   = agg + (long long)NN * D;

    const int feat4   = NN * D / 4;                     // 1.6M float4
    const int zblocks = (feat4 + 255) / 256;
    const int sblocks = (int)(((long long)NE * 16 + 255) / 256);
    const int cblocks = NN / 16;                        // 6250
    const int hblocks = (NN + 255) / 256;

    // ---- layer 1 ----
    zero4_kernel<<<zblocks, 256, 0, stream>>>((float4*)agg, feat4);
    scatter_add_kernel<<<sblocks, 256, 0, stream>>>(x, src, dst, agg);
    conv_wmma_kernel<<<cblocks, 32, 0, stream>>>(x, agg, W1_rel, W1_root,
                                                 b1_rel, h, 1);
    // ---- layer 2 (h updated in place: each wave owns its 16 rows) ----
    zero4_kernel<<<zblocks, 256, 0, stream>>>((float4*)agg, feat4);
    scatter_add_kernel<<<sblocks, 256, 0, stream>>>(h, src, dst, agg);
    conv_wmma_kernel<<<cblocks, 32, 0, stream>>>(h, agg, W2_rel, W2_root,
                                                 b2_rel, h, 1);
    // ---- head ----
    head_kernel<<<hblocks, 256, 0, stream>>>(h, W_lin, b_lin, out);
}